// Aggregator_50525995270157
// MI455X (gfx1250) — compile-verified
//
#include <hip/hip_runtime.h>
#include <hip/hip_bf16.h>
#include <math.h>

// ---------------------------------------------------------------------------
// GCNII-style aggregator for MI455X (gfx1250, wave32, WMMA + TDM)
//   hi   = ego + A_coo @ ego                      (atomic SpMM, L2 resident)
//   res  = 0.9*hi + 0.1*(h0 @ w_h0^T + b_h0)     (WMMA f32 GEMM, in-place)
//   emb1 = res @ ((1-beta) + beta*W)             (WMMA f32 GEMM, in-place)
//   out  = LN(LeakyReLU(emb1 @ w_lin^T + b_lin)) (WMMA f32 GEMM + reg LN)
// Matrix math: V_WMMA_F32_16X16X4_F32 (exact fp32). Tiles staged into LDS by
// the Tensor Data Mover (tensor_load_to_lds) with hardware LDS padding.
// ---------------------------------------------------------------------------

typedef __attribute__((ext_vector_type(2))) float    v2f;
typedef __attribute__((ext_vector_type(8))) float    v8f;
typedef __attribute__((ext_vector_type(4))) unsigned v4u;
typedef __attribute__((ext_vector_type(8))) int      v8i;
typedef __attribute__((ext_vector_type(4))) int      v4i;

#define DD 128
#define W_STRIDE  136   // weights: TDM pads 8 dwords per 128 -> bank 8j+k, <=2-way
#define AT_STRIDE 132   // A tiles: TDM pads 4 dwords per 128 -> b64-aligned frags
#define WAVES_PER_BLOCK 2
#define ROWS_PER_WAVE 16

__constant__ const float kAlpha   = 0.1f;
__constant__ const float kOneMinA = 0.9f;
// beta = log(LAMDA/LAYER + 1) = log(1.5)
__constant__ const float kBetaIM  = 0.4054651081081644f;

// --------------------------- SpMM (COO, atomics) ---------------------------
__global__ void k_spmm(const float* __restrict__ ego,
                       const float* __restrict__ vals,
                       const int*   __restrict__ rowi,
                       const int*   __restrict__ coli,
                       float* __restrict__ hi, int E) {
  const int gid  = blockIdx.x * blockDim.x + threadIdx.x;
  const int e    = gid >> 5;
  if (e >= E) return;
  const int lane = threadIdx.x & 31;
  const int c = coli[e];
  const int r = rowi[e];
  const float v = vals[e];
  const float4 x = *(const float4*)(ego + (size_t)c * DD + lane * 4);
  float* p = hi + (size_t)r * DD + lane * 4;
  atomicAdd(p + 0, v * x.x);
  atomicAdd(p + 1, v * x.y);
  atomicAdd(p + 2, v * x.z);
  atomicAdd(p + 3, v * x.w);
}

// ----------------------- TDM 2D tile load into LDS -------------------------
// D# per CDNA5 ISA: group0 = {flags, lds_addr, gaddr},
// group1 = dims/strides/pad, groups 2/3 zero (2D tensor). data_size = 4B.
// This toolchain's builtin is the 6-arg clang-23 form:
//   (uint32x4 g0, int32x8 g1, int32x4 g2, int32x4 g3, int32x8 g4, i32 cpol)
__device__ __forceinline__ int rfl(unsigned x) {
  return __builtin_amdgcn_readfirstlane((int)x);
}

__device__ __forceinline__ void tdm_load_2d(const float* gptr, unsigned lds_off,
                                            unsigned tile_rows,
                                            unsigned pad_amount_field) {
  const unsigned long long ga = (unsigned long long)(uintptr_t)gptr;
  v4u g0;
  g0.x = (unsigned)rfl(1u);                              // count=1, user desc
  g0.y = (unsigned)rfl(lds_off);                         // lds_addr (bytes)
  g0.z = (unsigned)rfl((unsigned)ga);                    // global_addr[31:0]
  g0.w = (unsigned)rfl((unsigned)(ga >> 32) | (2u << 30));  // addr hi | type=2
  v8i g1;
  // data_size=4B(2), pad_enable, pad_interval=6 (128 dw), pad_amount field
  g1[0] = rfl((2u << 16) | (1u << 20) | (6u << 22) | (pad_amount_field << 25));
  g1[1] = rfl((unsigned)DD << 16);        // tensor_dim0 = 128 (lo 16)
  g1[2] = rfl(tile_rows << 16);           // tensor_dim0 hi=0 | tensor_dim1 lo
  g1[3] = rfl((unsigned)DD << 16);        // tensor_dim1 hi=0 | tile_dim0 = 128
  g1[4] = rfl(tile_rows);                 // tile_dim1 | tile_dim2 = 0
  g1[5] = rfl((unsigned)DD);              // tensor_dim0_stride = 128
  g1[6] = 0;
  g1[7] = 0;
  const v4i z4 = {0, 0, 0, 0};
  const v8i z8 = {0, 0, 0, 0, 0, 0, 0, 0};
  __builtin_amdgcn_tensor_load_to_lds(g0, g1, z4, z4, z8, 0);
}

// ------------------------- WMMA fragment helpers ----------------------------
// A frag (16x4 f32): lane (idx,half): {A[idx][k0+2h], A[idx][k0+2h+1]}
// B frag (4x16 f32): lane (idx,half): {B[k0+2h][n0+idx], B[k0+2h+1][n0+idx]}
// C/D (16x16 f32):   reg r, lane (idx,half) -> element [8*half+r][n0+idx]

// GEMM vs transposed weight: out = A @ W^T, i.e. B[k][j] = W[j][k].
// W staged row-major (stride W_STRIDE) -> B frag is one aligned b64 from a row.
__device__ __forceinline__ void gemm_wt(v8f acc[8], const float* __restrict__ Ap,
                                        const float* __restrict__ Ws,
                                        int idx, int half) {
  #pragma unroll 2
  for (int k0 = 0; k0 < DD; k0 += 4) {
    const v2f a = *(const v2f*)(Ap + k0);
    #pragma unroll
    for (int nt = 0; nt < 8; ++nt) {
      const v2f b =
          *(const v2f*)(Ws + (nt * 16 + idx) * W_STRIDE + k0 + 2 * half);
      acc[nt] = __builtin_amdgcn_wmma_f32_16x16x4_f32(
          false, a, false, b, (short)0, acc[nt], false, false);
    }
  }
}

// GEMM vs identity-mapped weight: B[k][j] = (1-beta) + beta*W[k][j].
__device__ __forceinline__ void gemm_im(v8f acc[8], const float* __restrict__ Ap,
                                        const float* __restrict__ Ws,
                                        int idx, int half) {
  const float c1 = 1.0f - kBetaIM;
  #pragma unroll 2
  for (int k0 = 0; k0 < DD; k0 += 4) {
    const v2f a = *(const v2f*)(Ap + k0);
    #pragma unroll
    for (int nt = 0; nt < 8; ++nt) {
      v2f b;
      b.x = fmaf(kBetaIM, Ws[(k0 + 2 * half)     * W_STRIDE + nt * 16 + idx], c1);
      b.y = fmaf(kBetaIM, Ws[(k0 + 2 * half + 1) * W_STRIDE + nt * 16 + idx], c1);
      acc[nt] = __builtin_amdgcn_wmma_f32_16x16x4_f32(
          false, a, false, b, (short)0, acc[nt], false, false);
    }
  }
}

// Shared prologue: TDM-stage weights (wave 0) + this wave's 16x128 A tile.
__device__ __forceinline__ const float* stage_tiles(const float* __restrict__ wmat,
                                                    const float* __restrict__ act,
                                                    float* smem, int& rowbase,
                                                    int& idx, int& half) {
  const int tid = threadIdx.x;
  const int w = __builtin_amdgcn_readfirstlane(tid >> 5);
  const int lane = tid & 31;
  idx = lane & 15;
  half = lane >> 4;
  rowbase = (blockIdx.x * WAVES_PER_BLOCK + w) * ROWS_PER_WAVE;
  float* At_w = smem + DD * W_STRIDE + w * ROWS_PER_WAVE * AT_STRIDE;
  const unsigned lds_base = (unsigned)(uintptr_t)smem;   // LDS aperture low bits
  if (w == 0) {
    // 128x128 weight tile, padded to stride 136 dwords (pad field 7 -> 8 dw)
    tdm_load_2d(wmat, lds_base, DD, 7u);
  }
  // This wave's 16x128 activation tile, padded to stride 132 (pad field 3 -> 4 dw)
  tdm_load_2d(act + (size_t)rowbase * DD,
              (unsigned)(uintptr_t)At_w, ROWS_PER_WAVE, 3u);
  __builtin_amdgcn_s_wait_tensorcnt(0);
  __syncthreads();
  return At_w + idx * AT_STRIDE + 2 * half;
}

// ---------------- K2: hi = 0.9*hi + 0.1*(h0 @ w_h0^T + b_h0) ----------------
__global__ void k_residual(const float* __restrict__ h0,
                           const float* __restrict__ w_h0,
                           const float* __restrict__ b_h0,
                           float* __restrict__ hi) {
  extern __shared__ float smem[];
  int rowbase, idx, half;
  const float* Ap = stage_tiles(w_h0, h0, smem, rowbase, idx, half);
  v8f acc[8];
  #pragma unroll
  for (int nt = 0; nt < 8; ++nt) acc[nt] = (v8f)0.f;
  gemm_wt(acc, Ap, smem, idx, half);
  #pragma unroll
  for (int nt = 0; nt < 8; ++nt) {
    const int col = nt * 16 + idx;
    const float bb = b_h0[col];
    #pragma unroll
    for (int r = 0; r < 8; ++r) {
      float* p = hi + (size_t)(rowbase + 8 * half + r) * DD + col;
      *p = kOneMinA * (*p) + kAlpha * (acc[nt][r] + bb);
    }
  }
}

// ---------------- K3: hi = hi @ ((1-beta) + beta*W), in-place ---------------
__global__ void k_identity_map(const float* __restrict__ weight,
                               float* __restrict__ hi) {
  extern __shared__ float smem[];
  int rowbase, idx, half;
  const float* Ap = stage_tiles(weight, hi, smem, rowbase, idx, half);
  v8f acc[8];
  #pragma unroll
  for (int nt = 0; nt < 8; ++nt) acc[nt] = (v8f)0.f;
  gemm_im(acc, Ap, smem, idx, half);
  #pragma unroll
  for (int nt = 0; nt < 8; ++nt) {
    const int col = nt * 16 + idx;
    #pragma unroll
    for (int r = 0; r < 8; ++r)
      hi[(size_t)(rowbase + 8 * half + r) * DD + col] = acc[nt][r];
  }
}

// ------- K4: out = LayerNorm(LeakyReLU(hi @ w_lin^T + b_lin)) ---------------
__global__ void k_final(const float* __restrict__ hi,
                        const float* __restrict__ w_lin,
                        const float* __restrict__ b_lin,
                        const float* __restrict__ gamma,
                        const float* __restrict__ beta_ln,
                        float* __restrict__ out) {
  extern __shared__ float smem[];
  int rowbase, idx, half;
  const float* Ap = stage_tiles(w_lin, hi, smem, rowbase, idx, half);
  v8f acc[8];
  #pragma unroll
  for (int nt = 0; nt < 8; ++nt) acc[nt] = (v8f)0.f;
  gemm_wt(acc, Ap, smem, idx, half);

  // bias + LeakyReLU + per-row moments; lane (idx,half) owns rows 8*half+r,
  // cols 16*nt+idx -> each 16-lane half holds complete rows.
  float s1[8], s2[8];
  #pragma unroll
  for (int r = 0; r < 8; ++r) { s1[r] = 0.f; s2[r] = 0.f; }
  #pragma unroll
  for (int nt = 0; nt < 8; ++nt) {
    const float bb = b_lin[nt * 16 + idx];
    #pragma unroll
    for (int r = 0; r < 8; ++r) {
      float x = acc[nt][r] + bb;
      x = (x > 0.f) ? x : 0.01f * x;
      acc[nt][r] = x;
      s1[r] += x;
      s2[r] += x * x;
    }
  }
  #pragma unroll
  for (int off = 1; off < 16; off <<= 1) {   // masks <16 keep halves separate
    #pragma unroll
    for (int r = 0; r < 8; ++r) {
      s1[r] += __shfl_xor(s1[r], off);
      s2[r] += __shfl_xor(s2[r], off);
    }
  }
  float mean[8], rstd[8];
  #pragma unroll
  for (int r = 0; r < 8; ++r) {
    mean[r] = s1[r] * (1.0f / DD);
    const float var = s2[r] * (1.0f / DD) - mean[r] * mean[r];
    rstd[r] = rsqrtf(var + 1e-5f);
  }
  #pragma unroll
  for (int nt = 0; nt < 8; ++nt) {
    const int col = nt * 16 + idx;
    const float g = gamma[col], be = beta_ln[col];
    #pragma unroll
    for (int r = 0; r < 8; ++r) {
      out[(size_t)(rowbase + 8 * half + r) * DD + col] =
          (acc[nt][r] - mean[r]) * rstd[r] * g + be;
    }
  }
}

// ---------------------------------------------------------------------------
extern "C" void kernel_launch(void* const* d_in, const int* in_sizes, int n_in,
                              void* d_out, int out_size, void* d_ws, size_t ws_size,
                              hipStream_t stream) {
  const float* ego     = (const float*)d_in[0];
  const float* h0      = (const float*)d_in[1];
  const float* vals    = (const float*)d_in[2];
  const int*   rowi    = (const int*)d_in[3];
  const int*   coli    = (const int*)d_in[4];
  const float* weight  = (const float*)d_in[5];
  const float* w_h0    = (const float*)d_in[6];
  const float* b_h0    = (const float*)d_in[7];
  const float* w_lin   = (const float*)d_in[8];
  const float* b_lin   = (const float*)d_in[9];
  const float* gamma   = (const float*)d_in[10];
  const float* beta_ln = (const float*)d_in[11];
  float* out = (float*)d_out;
  float* hi  = (float*)d_ws;                 // N x 128 fp32 scratch (51.2 MB)

  const int N = in_sizes[0] / DD;            // 100000 (multiple of 32)
  const int E = in_sizes[2];                 // 1600000

  // hi = ego, then hi += A @ ego
  (void)hipMemcpyAsync(hi, ego, (size_t)N * DD * sizeof(float),
                       hipMemcpyDeviceToDevice, stream);
  {
    const long long threads = (long long)E * 32;
    const int blocks = (int)((threads + 255) / 256);
    k_spmm<<<blocks, 256, 0, stream>>>(ego, vals, rowi, coli, hi, E);
  }

  const int gemm_blocks = N / (WAVES_PER_BLOCK * ROWS_PER_WAVE);   // 3125 exact
  const size_t smem =
      (DD * W_STRIDE + WAVES_PER_BLOCK * ROWS_PER_WAVE * AT_STRIDE) * sizeof(float);

  k_residual<<<gemm_blocks, WAVES_PER_BLOCK * 32, smem, stream>>>(h0, w_h0, b_h0, hi);
  k_identity_map<<<gemm_blocks, WAVES_PER_BLOCK * 32, smem, stream>>>(weight, hi);
  k_final<<<gemm_blocks, WAVES_PER_BLOCK * 32, smem, stream>>>(hi, w_lin, b_lin,
                                                               gamma, beta_ln, out);
}